// GCNEncoder_3848290697675
// MI455X (gfx1250) — compile-verified
//
#include <hip/hip_runtime.h>

typedef __attribute__((ext_vector_type(2))) float v2f;
typedef __attribute__((ext_vector_type(8))) float v8f;

#define D 128          // feature dim (in == hid)
#define LDS_STRIDE 130 // padded stride for W^T in LDS (kills bank conflicts)

// ---------------------------------------------------------------------------
// degree / normalization
// ---------------------------------------------------------------------------
__global__ void init_deg_kernel(float* __restrict__ deg, int n) {
    int i = blockIdx.x * blockDim.x + threadIdx.x;
    if (i < n) deg[i] = 1.0f;  // self-loop contributes 1
}

__global__ void count_deg_kernel(const int* __restrict__ dst, int E,
                                 float* __restrict__ deg) {
    int e = blockIdx.x * blockDim.x + threadIdx.x;
    if (e < E) atomicAdd(&deg[dst[e]], 1.0f);
}

__global__ void dinv_kernel(float* __restrict__ deg, int n) {
    int i = blockIdx.x * blockDim.x + threadIdx.x;
    if (i < n) deg[i] = rsqrtf(deg[i]);  // deg >= 1 always (self-loop)
}

// ---------------------------------------------------------------------------
// GEMM: C[n,128] = A[n,128] @ W[128,128] via V_WMMA_F32_16X16X4_F32
// one wave computes a 16(M) x 128(N) strip; 4 waves / block -> 64 rows / block
// W is staged transposed (+pad) in LDS so B fragments are contiguous b64 loads
//
// Out-of-range rows: address is clamped to row 0 and loaded UNconditionally.
// C[M][:] depends only on A row M (lane M), and stores are guarded per-row,
// so garbage in clamped lanes never reaches memory. This keeps the inner loop
// free of exec-mask manipulation (pure load + wmma).
// ---------------------------------------------------------------------------
__global__ __launch_bounds__(128) void gemm_wmma_f32(
    const float* __restrict__ A, const float* __restrict__ W,
    float* __restrict__ C, int nrows) {
    __shared__ float wT[D * LDS_STRIDE];

    int tid = threadIdx.x;  // 0..127
    // stage W transposed: wT[col*LDS_STRIDE + row] = W[row*128 + col]
    for (int i = tid; i < D * D; i += 128) {
        int r = i >> 7, c = i & (D - 1);
        wT[c * LDS_STRIDE + r] = W[i];
    }
    __syncthreads();

    int wave = tid >> 5;
    int lane = tid & 31;
    int lrow = lane & 15;   // M (for A) / N (for B) position within lanes
    int lhi  = lane >> 4;   // 0: K=0..1 half, 1: K=2..3 half

    int m0 = blockIdx.x * 64 + wave * 16;

    v8f acc[8];
#pragma unroll
    for (int n = 0; n < 8; ++n) acc[n] = (v8f)(0.0f);

    int arow = m0 + lrow;
    int crow = arow < nrows ? arow : 0;  // clamp, load unconditionally
    const float* aptr = A + (size_t)crow * D + lhi * 2;

    for (int k = 0; k < D; k += 4) {
        // A fragment 16x4: lanes 0-15 hold K=k..k+1, lanes 16-31 K=k+2..k+3
        v2f a = *(const v2f*)(aptr + k);
#pragma unroll
        for (int n = 0; n < 8; ++n) {
            int col = n * 16 + lrow;
            // B fragment 4x16 (col-major source == contiguous in W^T)
            v2f b = *(const v2f*)(&wT[col * LDS_STRIDE + k + lhi * 2]);
            acc[n] = __builtin_amdgcn_wmma_f32_16x16x4_f32(
                false, a, false, b, (short)0, acc[n], false, false);
        }
    }

    // C layout: VGPR v holds M = v (lanes 0-15) / M = v+8 (lanes 16-31)
#pragma unroll
    for (int n = 0; n < 8; ++n) {
        int col = n * 16 + lrow;
#pragma unroll
        for (int v = 0; v < 8; ++v) {
            int row = m0 + v + 8 * lhi;
            if (row < nrows) C[(size_t)row * D + col] = acc[n][v];
        }
    }
}

// ---------------------------------------------------------------------------
// self-loop message (also initializes/overwrites the output buffer):
// out[i][:] = h[i][:] * dinv[i]^2        (one float4 per thread)
// ---------------------------------------------------------------------------
__global__ void self_loop_init_kernel(const float* __restrict__ h,
                                      const float* __restrict__ dinv,
                                      float* __restrict__ out, int n) {
    long long i = (long long)blockIdx.x * blockDim.x + threadIdx.x;
    long long total = (long long)n * (D / 4);
    if (i < total) {
        int node = (int)(i >> 5);  // 32 float4s per node
        float w = dinv[node] * dinv[node];
        float4 v = ((const float4*)h)[i];
        v.x *= w; v.y *= w; v.z *= w; v.w *= w;
        ((float4*)out)[i] = v;
    }
}

// ---------------------------------------------------------------------------
// edge scatter-add: one wave per edge, 32 lanes x float4 = 128 features
// out[dst] += h[src] * dinv[src]*dinv[dst]
// ---------------------------------------------------------------------------
__global__ __launch_bounds__(256) void edge_scatter_kernel(
    const int* __restrict__ src, const int* __restrict__ dst,
    const float* __restrict__ h, const float* __restrict__ dinv,
    float* __restrict__ out, int E) {
    int e = blockIdx.x * 8 + (threadIdx.x >> 5);
    int lane = threadIdx.x & 31;
    if (e >= E) return;
    int s = src[e];
    int d = dst[e];
    float w = dinv[s] * dinv[d];
    float4 v = ((const float4*)(h + (size_t)s * D))[lane];
    float* op = out + (size_t)d * D + lane * 4;
    atomicAdd(op + 0, v.x * w);
    atomicAdd(op + 1, v.y * w);
    atomicAdd(op + 2, v.z * w);
    atomicAdd(op + 3, v.w * w);
}

// ---------------------------------------------------------------------------
// bias (+ optional ReLU), in place
// ---------------------------------------------------------------------------
__global__ void bias_act_kernel(float* __restrict__ x,
                                const float* __restrict__ b,
                                long long total, int relu) {
    long long i = (long long)blockIdx.x * blockDim.x + threadIdx.x;
    if (i < total) {
        float v = x[i] + b[i & (D - 1)];
        if (relu) v = fmaxf(v, 0.0f);
        x[i] = v;
    }
}

// ---------------------------------------------------------------------------
// launch
// ---------------------------------------------------------------------------
extern "C" void kernel_launch(void* const* d_in, const int* in_sizes, int n_in,
                              void* d_out, int out_size, void* d_ws, size_t ws_size,
                              hipStream_t stream) {
    const float* x   = (const float*)d_in[0];
    const int*   ei  = (const int*)d_in[1];
    const float* W1  = (const float*)d_in[2];
    const float* b1  = (const float*)d_in[3];
    const float* W2  = (const float*)d_in[4];
    const float* b2  = (const float*)d_in[5];
    float* out = (float*)d_out;

    const int N = in_sizes[0] / D;      // 50000
    const int E = in_sizes[1] / 2;      // 640000
    const int* src = ei;
    const int* dst = ei + E;

    // workspace layout (floats): dinv[N] (padded) | bufA[N*D] | bufB[N*D]
    float* dinv = (float*)d_ws;
    size_t dinvPad = ((size_t)N + 63) & ~(size_t)63;
    float* bufA = dinv + dinvPad;               // h1, later h2
    float* bufB = bufA + (size_t)N * D;         // agg1 -> relu'd h

    const long long totalF  = (long long)N * D;
    const int nThreads256   = 256;
    const int nBlocksN      = (N + 255) / 256;
    const int nBlocksE      = (E + 255) / 256;
    const int nBlocksEdge   = (E + 7) / 8;                 // 8 edges / block
    const int nBlocksF4     = (int)(((long long)N * (D / 4) + 255) / 256);
    const int nBlocksF      = (int)((totalF + 255) / 256);
    const int nBlocksGemm   = (N + 63) / 64;

    // normalization: deg -> dinv (in place)
    init_deg_kernel<<<nBlocksN, nThreads256, 0, stream>>>(dinv, N);
    count_deg_kernel<<<nBlocksE, nThreads256, 0, stream>>>(dst, E, dinv);
    dinv_kernel<<<nBlocksN, nThreads256, 0, stream>>>(dinv, N);

    // ---- layer 1 ----
    gemm_wmma_f32<<<nBlocksGemm, 128, 0, stream>>>(x, W1, bufA, N);          // h1 = x @ W1
    self_loop_init_kernel<<<nBlocksF4, nThreads256, 0, stream>>>(bufA, dinv, bufB, N);
    edge_scatter_kernel<<<nBlocksEdge, nThreads256, 0, stream>>>(src, dst, bufA, dinv, bufB, E);
    bias_act_kernel<<<nBlocksF, nThreads256, 0, stream>>>(bufB, b1, totalF, 1);  // +b1, ReLU

    // ---- layer 2 ----
    gemm_wmma_f32<<<nBlocksGemm, 128, 0, stream>>>(bufB, W2, bufA, N);       // h2 = h @ W2
    self_loop_init_kernel<<<nBlocksF4, nThreads256, 0, stream>>>(bufA, dinv, out, N);
    edge_scatter_kernel<<<nBlocksEdge, nThreads256, 0, stream>>>(src, dst, bufA, dinv, out, E);
    bias_act_kernel<<<nBlocksF, nThreads256, 0, stream>>>(out, b2, totalF, 0);   // +b2
}